// Attention3D_30537217474647
// MI455X (gfx1250) — compile-verified
//
#include <hip/hip_runtime.h>
#include <hip/hip_bf16.h>
#include <math.h>

// ---------------------------------------------------------------------------
// Attention3D on gfx1250 using fp32 WMMA (V_WMMA_F32_16X16X4_F32).
//   x:[2,128,48,48,48] f32, w_qkv:[384,128], w_out:[128,128], b_out:[128]
//   NS = 48^3 = 110592 spatial positions per batch, WD = 48*48 = 2304.
// Pipeline:
//   K1: qkv = w_qkv @ x_channels           (ws: 2*384*NS f32 = ~340 MB)
//   K2: per (b,h,d): sim -> softmax -> attn@v, staged into d_out
//   K3: y = w_out @ stage + b_out, in-place on d_out
// ---------------------------------------------------------------------------

#define NS      110592   // 48*48*48
#define WD      2304     // 48*48
#define HROWS   48

typedef float v2f __attribute__((ext_vector_type(2)));
typedef float v8f __attribute__((ext_vector_type(8)));

__device__ __forceinline__ v8f wmma_f32(v2f a, v2f b, v8f c) {
    // D(16x16) = A(16x4) * B(4x16) + C, full fp32
    return __builtin_amdgcn_wmma_f32_16x16x4_f32(
        /*neg_a=*/false, a, /*neg_b=*/false, b,
        /*c_mod=*/(short)0, c, /*reuse_a=*/false, /*reuse_b=*/false);
}

// ------------------------- Kernel 1: QKV projection -------------------------
// grid: (NS/(16*8)=864, 384/16=24, 2), block: 256 (8 waves, 1 tile each)
__global__ __launch_bounds__(256)
void qkv_gemm_kernel(const float* __restrict__ x,
                     const float* __restrict__ w_qkv,
                     float* __restrict__ qkv) {
    const int wave = threadIdx.x >> 5;
    const int lane = threadIdx.x & 31;
    const int half = lane >> 4;     // selects K pair {0,1} vs {2,3}
    const int m16  = lane & 15;

    const int nBase = (blockIdx.x * 8 + wave) * 16;
    const int oBase = blockIdx.y * 16;
    const int b     = blockIdx.z;

    // A: w_qkv row for this lane's M, pre-offset by K-half
    const float* wrow = w_qkv + (size_t)(oBase + m16) * 128 + half * 2;
    // B: x column (channel-strided), this lane's N
    const float* xcol = x + (size_t)b * 128 * NS + nBase + m16;

    v8f acc = {};
    for (int k0 = 0; k0 < 128; k0 += 4) {
        v2f a = { wrow[k0], wrow[k0 + 1] };
        const int c0 = k0 + half * 2;
        v2f bb = { xcol[(size_t)c0 * NS], xcol[(size_t)(c0 + 1) * NS] };
        if (k0 + 8 < 128)
            __builtin_prefetch(xcol + (size_t)(k0 + 8) * NS, 0, 1);
        acc = wmma_f32(a, bb, acc);
    }

    float* outp = qkv + ((size_t)b * 384 + oBase) * NS + nBase;
    #pragma unroll
    for (int r = 0; r < 8; ++r)
        outp[(size_t)(r + 8 * half) * NS + m16] = acc[r];
}

// ------------------------- Kernel 2: attention ------------------------------
// grid: 256 blocks = (b,h,d) groups; block: 288 threads = 9 waves
__global__ __launch_bounds__(288)
void attn_kernel(const float* __restrict__ qkv,
                 float* __restrict__ stage) {
    __shared__ float smat[HROWS][HROWS];   // sim -> attn (9 KB)

    const int grp = blockIdx.x;            // b*128 + hc, hc = h*32 + d
    const int b   = grp >> 7;
    const int hc  = grp & 127;

    const int tid  = threadIdx.x;
    const int wave = tid >> 5;
    const int lane = tid & 31;
    const int half = lane >> 4;
    const int m16  = lane & 15;

    const float scale = 0.17677669529663687f;  // 32^-0.5

    const float* qbase = qkv + ((size_t)b * 384 + hc) * NS;
    const float* kbase = qkv + ((size_t)b * 384 + 128 + hc) * NS;
    const float* vbase = qkv + ((size_t)b * 384 + 256 + hc) * NS;

    // ---- phase 1: sim = (q*scale) . k^T, one 16x16 tile per wave ----
    {
        const int ti = wave / 3, tj = wave % 3;
        const float* qr = qbase + (size_t)(ti * 16 + m16) * WD + half * 2;
        const float* kr = kbase + (size_t)(tj * 16 + m16) * WD + half * 2;
        v8f acc = {};
        for (int n0 = 0; n0 < WD; n0 += 4) {
            v2f a  = { qr[n0], qr[n0 + 1] };
            v2f bb = { kr[n0], kr[n0 + 1] };
            acc = wmma_f32(a, bb, acc);
        }
        #pragma unroll
        for (int r = 0; r < 8; ++r)
            smat[ti * 16 + r + 8 * half][tj * 16 + m16] = acc[r] * scale;
    }
    __syncthreads();

    // ---- phase 2: row softmax over j (48 threads, one row each) ----
    if (tid < HROWS) {
        float m = smat[tid][0];
        #pragma unroll 4
        for (int j = 1; j < HROWS; ++j) m = fmaxf(m, smat[tid][j]);
        float s = 0.f;
        #pragma unroll 4
        for (int j = 0; j < HROWS; ++j) {
            float e = __expf(smat[tid][j] - m);
            smat[tid][j] = e;
            s += e;
        }
        float inv = 1.f / s;
        #pragma unroll 4
        for (int j = 0; j < HROWS; ++j) smat[tid][j] *= inv;
    }
    __syncthreads();

    // ---- phase 3: out = attn @ v  (3 i-tiles x 144 n-tiles = 432 tiles) ----
    float* obase = stage + ((size_t)b * 128 + hc) * NS;
    for (int t = wave; t < 3 * (WD / 16); t += 9) {
        const int iBase = (t % 3) * 16;
        const int nBase = (t / 3) * 16;
        v8f acc = {};
        #pragma unroll 4
        for (int j0 = 0; j0 < HROWS; j0 += 4) {
            const int jj = j0 + half * 2;
            v2f a  = { smat[iBase + m16][jj], smat[iBase + m16][jj + 1] };
            v2f bb = { vbase[(size_t)jj * WD + nBase + m16],
                       vbase[(size_t)(jj + 1) * WD + nBase + m16] };
            acc = wmma_f32(a, bb, acc);
        }
        #pragma unroll
        for (int r = 0; r < 8; ++r)
            obase[(size_t)(iBase + r + 8 * half) * WD + nBase + m16] = acc[r];
    }
}

// --------------------- Kernel 3: output projection (in-place) ---------------
// grid: (NS/16=6912, 2); block: 256 (8 waves -> 8 o-tiles of 16)
__global__ __launch_bounds__(256)
void proj_kernel(const float* __restrict__ w_out,
                 const float* __restrict__ b_out,
                 float* __restrict__ stage) {
    __shared__ float xt[128][16];          // channel tile (8 KB)

    const int pBase = blockIdx.x * 16;
    const int b     = blockIdx.y;

    const int tid  = threadIdx.x;
    const int wave = tid >> 5;
    const int lane = tid & 31;
    const int half = lane >> 4;
    const int m16  = lane & 15;

    // stage ALL channels for these 16 positions before any write (in-place safe)
    for (int idx = tid; idx < 128 * 16; idx += 256) {
        const int c  = idx >> 4;
        const int nn = idx & 15;
        xt[c][nn] = stage[((size_t)b * 128 + c) * NS + pBase + nn];
    }
    __syncthreads();

    const int oBase = wave * 16;
    const float* wrow = w_out + (size_t)(oBase + m16) * 128 + half * 2;

    v8f acc = {};
    #pragma unroll 4
    for (int k0 = 0; k0 < 128; k0 += 4) {
        const int c0 = k0 + half * 2;
        v2f a  = { wrow[k0], wrow[k0 + 1] };
        v2f bb = { xt[c0][m16], xt[c0 + 1][m16] };
        acc = wmma_f32(a, bb, acc);
    }

    #pragma unroll
    for (int r = 0; r < 8; ++r) {
        const int o = oBase + r + 8 * half;
        stage[((size_t)b * 128 + o) * NS + pBase + m16] = acc[r] + b_out[o];
    }
}

// ---------------------------------------------------------------------------
extern "C" void kernel_launch(void* const* d_in, const int* in_sizes, int n_in,
                              void* d_out, int out_size, void* d_ws, size_t ws_size,
                              hipStream_t stream) {
    const float* x     = (const float*)d_in[0];   // [2,128,48,48,48]
    const float* w_qkv = (const float*)d_in[1];   // [384,128]
    const float* w_out = (const float*)d_in[2];   // [128,128]
    const float* b_out = (const float*)d_in[3];   // [128]
    float* out = (float*)d_out;                   // [2,128,48,48,48]
    float* qkv = (float*)d_ws;                    // 2*384*NS f32 (~340 MB)

    // K1: QKV projection
    qkv_gemm_kernel<<<dim3(NS / 128, 384 / 16, 2), 256, 0, stream>>>(x, w_qkv, qkv);
    // K2: attention per (b, head, dh-channel), staged into d_out
    attn_kernel<<<dim3(256), 288, 0, stream>>>(qkv, out);
    // K3: output projection + bias, in-place on d_out
    proj_kernel<<<dim3(NS / 16, 2), 256, 0, stream>>>(w_out, b_out, out);
}